// DequantSiluAndMulQuant_73907797229881
// MI455X (gfx1250) — compile-verified
//
#include <hip/hip_runtime.h>
#include <math.h>

// ---------------------------------------------------------------------------
// DequantSiluAndMulQuant for MI455X (gfx1250, wave32)
//
// x: (8192, 22016) fp32  ->  q: (8192, 11008) per-token-quantized, scale: (8192,)
//
// One 256-thread block (8 wave32s) per row. Each lane keeps its 11 float4
// activation chunks in VGPRs (fully unrolled -> no scratch), so x is read from
// HBM exactly once even though quantization needs the row amax first.
// ---------------------------------------------------------------------------

typedef float f4 __attribute__((ext_vector_type(4)));

#define NTOK   8192
#define DHALF  11008
#define NVEC   (DHALF / 4)              // 2752 float4 per half-row
#define TPB    256                      // 8 waves (wave32)
#define NWAVES (TPB / 32)
#define MAXV   ((NVEC + TPB - 1) / TPB) // 11 float4 per lane

__global__ __launch_bounds__(TPB)
void silu_mul_quant_kernel(const float* __restrict__ x,
                           const float* __restrict__ gate_scale_p,
                           const float* __restrict__ up_scale_p,
                           float* __restrict__ out_q,
                           float* __restrict__ out_scale)
{
    const int row = blockIdx.x;
    const int tid = threadIdx.x;

    const float sg = *gate_scale_p;   // uniform -> s_load
    const float su = *up_scale_p;

    const f4* __restrict__ gptr = (const f4*)(x + (size_t)row * (2 * DHALF));
    const f4* __restrict__ uptr = gptr + NVEC;

    // Hint the up-half into flight early (lowers to global_prefetch_b8).
    __builtin_prefetch((const void*)(uptr + tid), 0, 0);

    // ---- Pass 1: stream-load, dequant, SiLU*up, track |act| max -----------
    f4 acc[MAXV];
    float amax = 0.0f;

#pragma unroll
    for (int j = 0; j < MAXV; ++j) {
        const int idx = tid + j * TPB;
        if (idx < NVEC) {
            f4 g = __builtin_nontemporal_load(gptr + idx);
            f4 u = __builtin_nontemporal_load(uptr + idx);
#pragma unroll
            for (int c = 0; c < 4; ++c) {
                const float gd  = g[c] * sg;
                const float ud  = u[c] * su;
                // silu(gd) = gd * sigmoid(gd); v_exp_f32 + v_rcp_f32 fast path
                const float sig = __builtin_amdgcn_rcpf(1.0f + __expf(-gd));
                const float a   = gd * sig * ud;
                g[c] = a;
                amax = fmaxf(amax, fabsf(a));
            }
            acc[j] = g;   // stays in VGPRs (constant indices, fully unrolled)
        }
    }

    // ---- Row-amax reduction: wave32 shuffle tree + LDS across 8 waves -----
#pragma unroll
    for (int off = 16; off > 0; off >>= 1)
        amax = fmaxf(amax, __shfl_xor(amax, off, 32));

    __shared__ float wred[NWAVES];
    __shared__ float s_inv;
    const int lane = tid & 31;
    const int wave = tid >> 5;
    if (lane == 0) wred[wave] = amax;
    __syncthreads();
    if (tid == 0) {
        float m = wred[0];
#pragma unroll
        for (int w = 1; w < NWAVES; ++w) m = fmaxf(m, wred[w]);
        out_scale[row] = m * (1.0f / 127.0f);          // per-token quant scale
        s_inv = (m > 0.0f) ? (127.0f / m) : 0.0f;
    }
    __syncthreads();
    const float inv = s_inv;

    // ---- Pass 2 (registers only): quantize + NT store ---------------------
    float* __restrict__ orow = out_q + (size_t)row * DHALF;
#pragma unroll
    for (int j = 0; j < MAXV; ++j) {
        const int idx = tid + j * TPB;
        if (idx < NVEC) {
            f4 a = acc[j];
            f4 q;
#pragma unroll
            for (int c = 0; c < 4; ++c) {
                float v = rintf(a[c] * inv);           // round-half-even, like jnp.round
                v = fminf(fmaxf(v, -128.0f), 127.0f);  // clip to int8 range
                q[c] = v;
            }
            __builtin_nontemporal_store(q, ((f4*)orow) + idx);
        }
    }
}

// ---------------------------------------------------------------------------
extern "C" void kernel_launch(void* const* d_in, const int* in_sizes, int n_in,
                              void* d_out, int out_size, void* d_ws, size_t ws_size,
                              hipStream_t stream)
{
    const float* x  = (const float*)d_in[0];           // (8192, 22016) fp32
    const float* sg = (const float*)d_in[1];           // scalar
    const float* su = (const float*)d_in[2];           // scalar

    float* out       = (float*)d_out;
    float* out_q     = out;                            // 8192*11008 values
    float* out_scale = out + (size_t)NTOK * DHALF;     // 8192 scales

    hipLaunchKernelGGL(silu_mul_quant_kernel,
                       dim3(NTOK), dim3(TPB), 0, stream,
                       x, sg, su, out_q, out_scale);
}